// PairUpdate_4337916970068
// MI455X (gfx1250) — compile-verified
//
#include <hip/hip_runtime.h>

typedef __attribute__((ext_vector_type(2))) float v2f;
typedef __attribute__((ext_vector_type(8))) float v8f;

#define LSEQ 1024
#define DM   128
#define DH   32
#define DP   64
#define LN_EPS 1e-5f

// ---------------------------------------------------------------------------
// Kernel 1: LayerNorm(s) then left = h@Wl + bl, right = h@Wr + br
// One block (128 threads = 4 waves) per sequence row.
// ---------------------------------------------------------------------------
__global__ __launch_bounds__(128) void ln_proj_kernel(
    const float* __restrict__ s,
    const float* __restrict__ gamma, const float* __restrict__ beta,
    const float* __restrict__ Wl, const float* __restrict__ bl,
    const float* __restrict__ Wr, const float* __restrict__ br,
    float* __restrict__ left, float* __restrict__ right)
{
  const int row = blockIdx.x;       // 0..1023
  const int t   = threadIdx.x;      // 0..127 (channel)
  __shared__ float h[DM];
  __shared__ float red[4];

  float x = s[row * DM + t];

  // mean: wave32 shuffle reduce, then 4-wave combine via LDS
  float v = x;
  #pragma unroll
  for (int off = 16; off > 0; off >>= 1) v += __shfl_down(v, off, 32);
  if ((t & 31) == 0) red[t >> 5] = v;
  __syncthreads();
  float mu = (red[0] + red[1] + red[2] + red[3]) * (1.0f / DM);
  __syncthreads();

  float d = x - mu;
  v = d * d;
  #pragma unroll
  for (int off = 16; off > 0; off >>= 1) v += __shfl_down(v, off, 32);
  if ((t & 31) == 0) red[t >> 5] = v;
  __syncthreads();
  float var = (red[0] + red[1] + red[2] + red[3]) * (1.0f / DM);

  h[t] = d * rsqrtf(var + LN_EPS) * gamma[t] + beta[t];
  __syncthreads();

  // projections: threads 0..31 -> left, 32..63 -> right
  if (t < DH) {
    float acc = bl[t];
    #pragma unroll 4
    for (int c = 0; c < DM; ++c) acc += h[c] * Wl[c * DH + t];
    left[row * DH + t] = acc;
  } else if (t < 2 * DH) {
    const int dd = t - DH;
    float acc = br[dd];
    #pragma unroll 4
    for (int c = 0; c < DM; ++c) acc += h[c] * Wr[c * DH + dd];
    right[row * DH + dd] = acc;
  }
}

// ---------------------------------------------------------------------------
// Kernel 2: rW[j,d,p] = sum_c right[j,c] * Wo[(d*DH + c)*DP + p]
// GEMM M=1024 (j), N=2048 (d,p), K=32.  One wave per 16x16 tile,
// K swept as 8x V_WMMA_F32_16X16X4_F32.
// ---------------------------------------------------------------------------
__global__ __launch_bounds__(256) void rw_kernel(
    const float* __restrict__ right, const float* __restrict__ Wo,
    float* __restrict__ rW)
{
  const int wid  = threadIdx.x >> 5;
  const int lane = threadIdx.x & 31;
  const int m    = lane & 15;
  const int hi   = lane >> 4;            // 0 or 1

  const int tile = blockIdx.x * 8 + wid; // 0..8191
  const int nt   = tile & 127;           // 128 n-tiles of 16
  const int jt   = tile >> 7;            // 0..63
  const int dd   = nt >> 2;              // 0..31
  const int p0   = (nt & 3) * 16;
  const int j0   = jt * 16;

  const float* arow = &right[(j0 + m) * DH];
  v8f c = {};
  #pragma unroll
  for (int kb = 0; kb < DH; kb += 4) {
    const int k = kb + 2 * hi;
    // A 16x4 f32: lane m holds row M=m; VGPR0/1 = K=k, k+1
    v2f a = *(const v2f*)(arow + k);
    // B 4x16 f32: lane m holds col N=m; VGPR0/1 = K=k, k+1
    v2f b;
    b.x = Wo[(dd * DH + k)     * DP + p0 + m];
    b.y = Wo[(dd * DH + k + 1) * DP + p0 + m];
    c = __builtin_amdgcn_wmma_f32_16x16x4_f32(false, a, false, b,
                                              (short)0, c, false, false);
  }
  #pragma unroll
  for (int r = 0; r < 8; ++r) {
    const int j = j0 + r + 8 * hi;       // C/D: VGPR r -> row r + 8*hi, col m
    rW[j * (DH * DP) + dd * DP + p0 + m] = c[r];
  }
}

// ---------------------------------------------------------------------------
// Kernel 3: out[i,j,p] = pair[i,j,p] + bo[p] + sum_d left[i,d]*rW[j,d,p]
// One wave per (16-row i-tile, j); 4 p-tiles of 16 held in 4 accumulators
// (32 WMMA per wave, A reused 4x).  pair/out streamed non-temporally so the
// 512 MB stream does not evict rW/left from L2 (192 MB).
// ---------------------------------------------------------------------------
__global__ __launch_bounds__(256) void pair_update_kernel(
    const float* __restrict__ left, const float* __restrict__ rW,
    const float* __restrict__ pair, const float* __restrict__ bo,
    float* __restrict__ out)
{
  const int wid  = threadIdx.x >> 5;
  const int lane = threadIdx.x & 31;
  const int m    = lane & 15;
  const int hi   = lane >> 4;

  const int tile = blockIdx.x * 8 + wid;  // 0..65535
  const int j    = tile & (LSEQ - 1);
  const int it   = tile >> 10;            // 0..63
  const int i0   = it * 16;

  v8f acc[4];
  #pragma unroll
  for (int pt = 0; pt < 4; ++pt) acc[pt] = (v8f){};

  const float* arow = &left[(i0 + m) * DH];
  const float* brow = &rW[j * (DH * DP)];

  #pragma unroll
  for (int kb = 0; kb < DH; kb += 4) {
    const int k = kb + 2 * hi;
    v2f a = *(const v2f*)(arow + k);
    #pragma unroll
    for (int pt = 0; pt < 4; ++pt) {
      v2f b;
      b.x = brow[k       * DP + pt * 16 + m];
      b.y = brow[(k + 1) * DP + pt * 16 + m];
      acc[pt] = __builtin_amdgcn_wmma_f32_16x16x4_f32(false, a, false, b,
                                                      (short)0, acc[pt],
                                                      false, false);
    }
  }

  #pragma unroll
  for (int pt = 0; pt < 4; ++pt) {
    const float bop = bo[pt * 16 + m];
    #pragma unroll
    for (int r = 0; r < 8; ++r) {
      const int idx = (i0 + r + 8 * hi) * (LSEQ * DP) + j * DP + pt * 16 + m;
      const float pv = __builtin_nontemporal_load(&pair[idx]);
      __builtin_nontemporal_store(pv + bop + acc[pt][r], &out[idx]);
    }
  }
}

// ---------------------------------------------------------------------------
extern "C" void kernel_launch(void* const* d_in, const int* in_sizes, int n_in,
                              void* d_out, int out_size, void* d_ws, size_t ws_size,
                              hipStream_t stream) {
  const float* s     = (const float*)d_in[0];
  const float* pair  = (const float*)d_in[1];
  const float* gamma = (const float*)d_in[2];
  const float* beta  = (const float*)d_in[3];
  const float* Wl    = (const float*)d_in[4];
  const float* bl    = (const float*)d_in[5];
  const float* Wr    = (const float*)d_in[6];
  const float* br    = (const float*)d_in[7];
  const float* Wo    = (const float*)d_in[8];
  const float* bo    = (const float*)d_in[9];
  float* out = (float*)d_out;

  float* left  = (float*)d_ws;                  // 1024*32 floats
  float* right = left  + LSEQ * DH;             // 1024*32 floats
  float* rW    = right + LSEQ * DH;             // 1024*2048 floats (8 MB)

  ln_proj_kernel<<<LSEQ, 128, 0, stream>>>(s, gamma, beta, Wl, bl, Wr, br,
                                           left, right);
  rw_kernel<<<1024, 256, 0, stream>>>(right, Wo, rW);
  pair_update_kernel<<<8192, 256, 0, stream>>>(left, rW, pair, bo, out);
}